// ScheduledSamplingGRULayer_87677462381239
// MI455X (gfx1250) — compile-verified
//
#include <hip/hip_runtime.h>

typedef __attribute__((ext_vector_type(2))) float v2f;
typedef __attribute__((ext_vector_type(8))) float v8f;

#define B_  64
#define T_  2048
#define H_  256
#define D_  32
#define SS_ 1024
#define G3  (3 * H_)   // 768

// LDS row strides (floats): rows land on distinct banks; every
// ds_load_b64 / async b128 destination stays 8/16-byte aligned.
#define XS_STRIDE 260   // 260*4 = 1040 B, %16==0, bank stride 4
#define P1_STRIDE 36    // 36*4  = 144 B,  %16==0

__device__ __forceinline__ float sigmoidf_(float v) {
  return 1.0f / (1.0f + __expf(-v));
}

// D = A(16x4 f32) * B(4x16 f32) + C(16x16 f32), wave32 WMMA.
__device__ __forceinline__ v8f wmma_f32(v2f a, v2f b, v8f c) {
  return __builtin_amdgcn_wmma_f32_16x16x4_f32(
      /*neg_a=*/false, a, /*neg_b=*/false, b,
      /*c_mod=*/(short)0, c, /*reuse_a=*/false, /*reuse_b=*/false);
}

// Async DMA of 16 bytes/lane straight into LDS (no VGPR round-trip).
// dsaddr = LDS byte offset (low 32 bits of the flat shared address).
__device__ __forceinline__ void async_g2lds_b128(unsigned lds_off,
                                                 const float* gaddr) {
  asm volatile("global_load_async_to_lds_b128 %0, %1, off"
               :: "v"(lds_off), "v"(gaddr)
               : "memory");
}

__device__ __forceinline__ void wait_asynccnt0() {
  asm volatile("s_wait_asynccnt 0x0" ::: "memory");
}

// ---------------------------------------------------------------------------
// Kernel 1: fused  pred1 = x@Wd+bd ; gates from (pred1@Wk+bi) and (x@Uk+br);
// gru = z*x + (1-z)*tanh(xh + r*hh).  One block = 16 flat (b,t) rows.
// Writes pred1 into preds[] for ALL t (kernel 2 overwrites t >= SS+1).
// ---------------------------------------------------------------------------
__global__ __launch_bounds__(256) void gru_fused_kernel(
    const float* __restrict__ x,  const float* __restrict__ Wd,
    const float* __restrict__ bd, const float* __restrict__ Wk,
    const float* __restrict__ Uk, const float* __restrict__ bi,
    const float* __restrict__ br, float* __restrict__ preds,
    float* __restrict__ gru) {
  __shared__ float xs[16 * XS_STRIDE + 16 * P1_STRIDE];
  float* p1s = xs + 16 * XS_STRIDE;

  const int row0 = blockIdx.x * 16;   // flat row index into [B*T]
  const int tid  = threadIdx.x;
  const int wave = tid >> 5;
  const int lane = tid & 31;
  const int lm   = lane & 15;         // M (A/C) or N (B/C) low index
  const int lh   = lane >> 4;         // half selector

  // Stage x tile [16 x 256] into LDS via async global->LDS DMA (B128/lane).
  // 16 rows x 64 float4 = 1024 chunks; 256 threads -> 4 async issues each.
  {
    const float* xg = x + (size_t)row0 * H_;
    #pragma unroll
    for (int it = 0; it < 4; ++it) {
      int i = tid + it * 256;
      int r = i >> 6, c = i & 63;                    // row, float4-column
      unsigned lds_off =
          (unsigned)(uintptr_t)(&xs[r * XS_STRIDE + c * 4]);  // addr[31:0]
      async_g2lds_b128(lds_off, xg + (size_t)r * H_ + c * 4);
    }
    wait_asynccnt0();
  }
  __syncthreads();

  // pred1: waves 0,1 each own one 16x16 N-tile of D=32.
  if (wave < 2) {
    const int n0 = wave * 16;
    float bias = bd[n0 + lm];
    v8f acc;
    #pragma unroll
    for (int i = 0; i < 8; ++i) acc[i] = bias;
    #pragma unroll 8
    for (int k = 0; k < H_; k += 4) {
      int ka = k + 2 * lh;
      v2f a = *(const v2f*)&xs[lm * XS_STRIDE + ka];
      v2f b;
      b.x = Wd[(ka + 0) * D_ + n0 + lm];
      b.y = Wd[(ka + 1) * D_ + n0 + lm];
      acc = wmma_f32(a, b, acc);
    }
    #pragma unroll
    for (int i = 0; i < 8; ++i) {
      int m = i + 8 * lh;
      p1s[m * P1_STRIDE + n0 + lm]             = acc[i];
      preds[(size_t)(row0 + m) * D_ + n0 + lm] = acc[i];
    }
  }
  __syncthreads();

  // Gates: 16 column-tiles of H=256; each of the 8 waves handles 2.
  for (int cti = 0; cti < 2; ++cti) {
    const int j0 = (wave * 2 + cti) * 16;
    const int jc = j0 + lm;   // column within H (valid for both lane halves)
    v8f accZ, accR, accXH, accHH;
    {
      float vz  = bi[jc]          + br[jc];          // z gets gx+gh fused
      float vr  = bi[H_ + jc]     + br[H_ + jc];     // r gets gx+gh fused
      float vxh = bi[2 * H_ + jc];
      float vhh = br[2 * H_ + jc];
      #pragma unroll
      for (int i = 0; i < 8; ++i) {
        accZ[i] = vz; accR[i] = vr; accXH[i] = vxh; accHH[i] = vhh;
      }
    }
    // gx part: pred1 @ Wk  (K = 32)
    #pragma unroll
    for (int k = 0; k < D_; k += 4) {
      int ka = k + 2 * lh;
      v2f a = *(const v2f*)&p1s[lm * P1_STRIDE + ka];
      v2f bz, brr, bh;
      bz.x  = Wk[(ka + 0) * G3 + jc];           bz.y  = Wk[(ka + 1) * G3 + jc];
      brr.x = Wk[(ka + 0) * G3 + H_ + jc];      brr.y = Wk[(ka + 1) * G3 + H_ + jc];
      bh.x  = Wk[(ka + 0) * G3 + 2 * H_ + jc];  bh.y  = Wk[(ka + 1) * G3 + 2 * H_ + jc];
      accZ  = wmma_f32(a, bz,  accZ);
      accR  = wmma_f32(a, brr, accR);
      accXH = wmma_f32(a, bh,  accXH);
    }
    // gh part: x @ Uk  (K = 256) — dominant GEMM, weights L2-resident.
    #pragma unroll 4
    for (int k = 0; k < H_; k += 4) {
      int ka = k + 2 * lh;
      v2f a = *(const v2f*)&xs[lm * XS_STRIDE + ka];
      v2f bz, brr, bh;
      bz.x  = Uk[(ka + 0) * G3 + jc];           bz.y  = Uk[(ka + 1) * G3 + jc];
      brr.x = Uk[(ka + 0) * G3 + H_ + jc];      brr.y = Uk[(ka + 1) * G3 + H_ + jc];
      bh.x  = Uk[(ka + 0) * G3 + 2 * H_ + jc];  bh.y  = Uk[(ka + 1) * G3 + 2 * H_ + jc];
      accZ  = wmma_f32(a, bz,  accZ);
      accR  = wmma_f32(a, brr, accR);
      accHH = wmma_f32(a, bh,  accHH);
    }
    // Elementwise gates + store gru_states (16 consecutive floats per vgpr
    // across lanes 0-15 / 16-31 -> coalesced 64B stores).
    #pragma unroll
    for (int i = 0; i < 8; ++i) {
      int m   = i + 8 * lh;
      float h = xs[m * XS_STRIDE + jc];
      float z = sigmoidf_(accZ[i]);
      float r = sigmoidf_(accR[i]);
      float hc = tanhf(accXH[i] + r * accHH[i]);
      gru[(size_t)(row0 + m) * H_ + jc] = z * h + (1.0f - z) * hc;
    }
  }
}

// ---------------------------------------------------------------------------
// Kernel 2: preds[b, t] = gru[b, t-1] @ Wd + bd  for t in [SS+1, T).
// Block = 16 t-rows for one batch; 2 waves cover the two 16-col tiles of D.
// ---------------------------------------------------------------------------
__global__ __launch_bounds__(64) void pred2_kernel(
    const float* __restrict__ gru, const float* __restrict__ Wd,
    const float* __restrict__ bd, float* __restrict__ preds) {
  const int b      = blockIdx.x >> 6;
  const int tile   = blockIdx.x & 63;
  const int ttBase = SS_ + 1 + tile * 16;
  const int wave   = threadIdx.x >> 5;
  const int lane   = threadIdx.x & 31;
  const int lm = lane & 15, lh = lane >> 4;
  const int n0 = wave * 16;

  // A-fragment row owned by this lane (clamped at the ragged tail).
  int tt  = ttBase + lm;
  int src = (tt - 1 < T_) ? (tt - 1) : (T_ - 1);
  const float* arow = gru + ((size_t)b * T_ + src) * H_;

  v8f acc;
  {
    float bias = bd[n0 + lm];
    #pragma unroll
    for (int i = 0; i < 8; ++i) acc[i] = bias;
  }
  #pragma unroll 8
  for (int k = 0; k < H_; k += 4) {
    int ka = k + 2 * lh;
    v2f a = *(const v2f*)&arow[ka];
    v2f bb;
    bb.x = Wd[(ka + 0) * D_ + n0 + lm];
    bb.y = Wd[(ka + 1) * D_ + n0 + lm];
    acc = wmma_f32(a, bb, acc);
  }
  #pragma unroll
  for (int i = 0; i < 8; ++i) {
    int m = i + 8 * lh;
    int t = ttBase + m;
    if (t < T_)
      preds[((size_t)b * T_ + t) * D_ + n0 + lm] = acc[i];
  }
}

extern "C" void kernel_launch(void* const* d_in, const int* in_sizes, int n_in,
                              void* d_out, int out_size, void* d_ws, size_t ws_size,
                              hipStream_t stream) {
  (void)in_sizes; (void)n_in; (void)out_size; (void)d_ws; (void)ws_size;
  const float* x  = (const float*)d_in[0];  // [B,T,H]
  const float* Wd = (const float*)d_in[1];  // [H,D]
  const float* bd = (const float*)d_in[2];  // [D]
  const float* Wk = (const float*)d_in[3];  // [D,3H]
  const float* Uk = (const float*)d_in[4];  // [H,3H]
  const float* bi = (const float*)d_in[5];  // [3H]
  const float* br = (const float*)d_in[6];  // [3H]

  float* preds = (float*)d_out;                       // [B,T,D]
  float* gru   = preds + (size_t)B_ * T_ * D_;        // [B,T,H]

  gru_fused_kernel<<<(B_ * T_) / 16, 256, 0, stream>>>(x, Wd, bd, Wk, Uk, bi,
                                                       br, preds, gru);
  pred2_kernel<<<B_ * 64, 64, 0, stream>>>(gru, Wd, bd, preds);
}